// DeltaDequantization_6236292513856
// MI455X (gfx1250) — compile-verified
//
#include <hip/hip_runtime.h>

typedef __attribute__((ext_vector_type(2))) float v2f;
typedef __attribute__((ext_vector_type(8))) float v8f;

#define ADAPT_STEP 32
#define NB        32
#define TSTEPS    2048
#define NCHUNKS   64          // TSTEPS / ADAPT_STEP
#define THREADS   256
#define ROWS_PER_WAVE 256     // 8 waves * 256 rows = 2048 rows per workgroup
#define TILES_PER_WAVE 16     // 16 rows per WMMA tile

// out[b,t] = pred_c + S_c * (x[b,t,:] . quant_bins)
// where per-chunk scalars come from a tiny sequential scan:
//   m_c = mean_s(x . quant_bins),  r_c = mean_s(x . change_scales)
//   pred_{c+1} = pred_c + S_c * m_c ;  S_{c+1} = S_c * r_c ; S_0=1, pred_0=0
__global__ __launch_bounds__(THREADS)
void delta_dequant_kernel(const float* __restrict__ x,
                          const float* __restrict__ qb,
                          const float* __restrict__ cs,
                          float* __restrict__ out)
{
    __shared__ float s_dA[TSTEPS];
    __shared__ float s_dC[TSTEPS];
    __shared__ float s_pred[NCHUNKS];   // stage B: m_c   -> stage C: pred_c
    __shared__ float s_scl[NCHUNKS];    // stage B: r_c   -> stage C: S_c

    const int b     = blockIdx.x;
    const int tid   = threadIdx.x;
    const int wave  = tid >> 5;         // wave32
    const int lane  = tid & 31;
    const int n     = lane & 15;        // column index of B/D fragment
    const int khalf = (lane >> 4) * 2;  // K sub-offset held by this half-wave

    // ---- Build 8 constant B fragments (4x16 f32): col0 = quant_bins, col1 = change_scales
    v2f bfrag[8];
#pragma unroll
    for (int k4 = 0; k4 < 8; ++k4) {
        const int kidx = k4 * 4 + khalf;
        float b0 = 0.0f, b1 = 0.0f;
        if (n == 0)      { b0 = qb[kidx]; b1 = qb[kidx + 1]; }
        else if (n == 1) { b0 = cs[kidx]; b1 = cs[kidx + 1]; }
        v2f bf; bf.x = b0; bf.y = b1;
        bfrag[k4] = bf;
    }

    const float* xb = x + (size_t)b * TSTEPS * NB;

    // ---- Stage A: WMMA dot products for this wave's 256 rows
    for (int tile = 0; tile < TILES_PER_WAVE; ++tile) {
        const int r0 = wave * ROWS_PER_WAVE + tile * 16;
        const float* rowp = xb + (size_t)(r0 + n) * NB + khalf;

        if (tile + 1 < TILES_PER_WAVE)
            __builtin_prefetch(xb + (size_t)(r0 + 16 + n) * NB, 0, 0); // global_prefetch_b8

        v8f acc = {};
#pragma unroll
        for (int k4 = 0; k4 < 8; ++k4) {
            v2f a = *(const v2f*)(rowp + k4 * 4);   // A[m][khalf..khalf+1], K-slice k4*4
            acc = __builtin_amdgcn_wmma_f32_16x16x4_f32(
                /*neg_a=*/false, a, /*neg_b=*/false, bfrag[k4],
                /*c_mod=*/(short)0, acc, /*reuse_a=*/false, /*reuse_b=*/false);
        }

        // D layout: VGPR j, lane l -> (M = j + (l>=16)*8, N = l&15).
        // Column 0 = dA, column 1 = dC -> lanes 0,1,16,17 hold everything.
        if (n <= 1) {
            float* dst = (n == 0) ? s_dA : s_dC;
            const int mbase = r0 + (lane >> 4) * 8;
#pragma unroll
            for (int j = 0; j < 8; ++j) dst[mbase + j] = acc[j];
        }
    }
    __syncthreads();

    // ---- Stage B: per-chunk means (64 chunks handled by threads 0..63)
    if (tid < NCHUNKS) {
        float sa = 0.0f, sc = 0.0f;
#pragma unroll
        for (int s = 0; s < ADAPT_STEP; ++s) {
            sa += s_dA[tid * ADAPT_STEP + s];
            sc += s_dC[tid * ADAPT_STEP + s];
        }
        s_pred[tid] = sa * (1.0f / ADAPT_STEP);  // m_c
        s_scl[tid]  = sc * (1.0f / ADAPT_STEP);  // r_c
    }
    __syncthreads();

    // ---- sequential 64-step scalar scan (one thread; trivial cost)
    if (tid == 0) {
        float pred = 0.0f, S = 1.0f;
        for (int c = 0; c < NCHUNKS; ++c) {
            const float m = s_pred[c];
            const float r = s_scl[c];
            s_pred[c] = pred;
            s_scl[c]  = S;
            pred += S * m;
            S    *= r;
        }
    }
    __syncthreads();

    // ---- Stage C: out[t] = pred_c + S_c * dA[t], 8 consecutive t per thread
    float* outb = out + (size_t)b * TSTEPS;
    const int t0 = tid * 8;               // 8 | 32 -> single chunk per thread
    const int c  = t0 >> 5;
    const float pred = s_pred[c];
    const float S    = s_scl[c];

    float4 o0, o1;
    o0.x = pred + S * s_dA[t0 + 0];
    o0.y = pred + S * s_dA[t0 + 1];
    o0.z = pred + S * s_dA[t0 + 2];
    o0.w = pred + S * s_dA[t0 + 3];
    o1.x = pred + S * s_dA[t0 + 4];
    o1.y = pred + S * s_dA[t0 + 5];
    o1.z = pred + S * s_dA[t0 + 6];
    o1.w = pred + S * s_dA[t0 + 7];
    *(float4*)(outb + t0)     = o0;       // global_store_b128
    *(float4*)(outb + t0 + 4) = o1;       // global_store_b128
}

extern "C" void kernel_launch(void* const* d_in, const int* in_sizes, int n_in,
                              void* d_out, int out_size, void* d_ws, size_t ws_size,
                              hipStream_t stream) {
    (void)n_in; (void)d_ws; (void)ws_size; (void)in_sizes;
    const float* x  = (const float*)d_in[0];   // (B, T, NB) f32
    const float* qb = (const float*)d_in[1];   // (NB,) f32
    const float* cs = (const float*)d_in[2];   // (NB,1) f32
    float* out = (float*)d_out;                // (B, T) f32

    const int B = out_size / TSTEPS;           // 1024
    delta_dequant_kernel<<<dim3(B), dim3(THREADS), 0, stream>>>(x, qb, cs, out);
}